// NumericalFeatureEmbedding_2293512536096
// MI455X (gfx1250) — compile-verified
//
#include <hip/hip_runtime.h>

typedef __attribute__((ext_vector_type(2))) float v2f;
typedef __attribute__((ext_vector_type(8))) float v8f;

// Problem sizes (fixed by the reference harness)
constexpr int Bsz = 16384;
constexpr int F   = 128;
constexpr int H   = 16;
constexpr int E   = 32;

// One wave32 per (16-batch-row tile, feature).
// Block = 256 threads = 8 waves = one batch tile x 8 features.
__global__ __launch_bounds__(256) void nfe_wmma_kernel(
    const float* __restrict__ x,    // [B, F]
    const float* __restrict__ W1,   // [F, H]
    const float* __restrict__ b1,   // [F, H]
    const float* __restrict__ W2,   // [F, H, E]
    const float* __restrict__ b2,   // [F, E]
    float* __restrict__ out)        // [B, F, E]
{
    const int lane = threadIdx.x & 31;
    const int wave = threadIdx.x >> 5;

    const int b0 = blockIdx.x * 16;            // batch tile start
    const int f  = blockIdx.y * 8 + wave;      // feature for this wave

    const int m  = lane & 15;                  // row within tile (A / C)
    const int n  = lane & 15;                  // col within tile (B / C)
    const int hi = lane >> 4;                  // lane half

    // ---- Build A operand: h[16 x 16] = relu(x ⊗ W1 + b1) for this feature.
    // A (16x4 f32) layout: VGPR j, lanes 0-15 hold K=j, lanes 16-31 hold K=j+2.
    // K-chunk kc covers K = 4*kc .. 4*kc+3  ->  k = 4*kc + 2*hi + j.
    const float xv = x[(b0 + m) * F + f];

    v2f a[4];
#pragma unroll
    for (int kc = 0; kc < 4; ++kc) {
#pragma unroll
        for (int j = 0; j < 2; ++j) {
            const int k = 4 * kc + 2 * hi + j;
            const float hval = __builtin_fmaf(xv, W1[f * H + k], b1[f * H + k]);
            a[kc][j] = hval > 0.0f ? hval : 0.0f;
        }
    }

    // ---- Load B operand: W2[f] as two 16x(4x4) K-chunked N-tiles (E = 2*16).
    // B (4x16 f32) layout mirrors A: VGPR j, lanes 0-15 K=j, lanes 16-31 K=j+2.
    v2f bmat[2][4];
#pragma unroll
    for (int t = 0; t < 2; ++t) {
#pragma unroll
        for (int kc = 0; kc < 4; ++kc) {
#pragma unroll
            for (int j = 0; j < 2; ++j) {
                const int k = 4 * kc + 2 * hi + j;
                bmat[t][kc][j] = W2[(f * H + k) * E + t * 16 + n];
            }
        }
    }

    // ---- Accumulate D = A x B (+bias) with chained FP32 WMMA, K=16 in 4 steps.
#pragma unroll
    for (int t = 0; t < 2; ++t) {
        v8f acc = {0.f, 0.f, 0.f, 0.f, 0.f, 0.f, 0.f, 0.f};
#pragma unroll
        for (int kc = 0; kc < 4; ++kc) {
            // (neg_a, A, neg_b, B, c_mod, C, reuse_a, reuse_b)
            acc = __builtin_amdgcn_wmma_f32_16x16x4_f32(
                false, a[kc], false, bmat[t][kc], (short)0, acc, false, false);
        }

        const float bias = b2[f * E + t * 16 + n];

        // C/D layout: VGPR v -> row (v + 8*hi), col n.
#pragma unroll
        for (int v = 0; v < 8; ++v) {
            const int row = b0 + v + 8 * hi;
            out[(row * F + f) * E + t * 16 + n] = acc[v] + bias;
        }
    }
}

extern "C" void kernel_launch(void* const* d_in, const int* in_sizes, int n_in,
                              void* d_out, int out_size, void* d_ws, size_t ws_size,
                              hipStream_t stream) {
    (void)in_sizes; (void)n_in; (void)out_size; (void)d_ws; (void)ws_size;

    const float* x  = (const float*)d_in[0];
    const float* W1 = (const float*)d_in[1];
    const float* b1 = (const float*)d_in[2];
    const float* W2 = (const float*)d_in[3];
    const float* b2 = (const float*)d_in[4];
    float* out      = (float*)d_out;

    dim3 grid(Bsz / 16, F / 8);   // (1024, 16)
    dim3 block(256);
    nfe_wmma_kernel<<<grid, block, 0, stream>>>(x, W1, b1, W2, b2, out);
}